// ColorConsistencyLoss_84387517432496
// MI455X (gfx1250) — compile-verified
//
#include <hip/hip_runtime.h>

typedef float v2f __attribute__((ext_vector_type(2)));
typedef float v8f __attribute__((ext_vector_type(8)));

namespace {
constexpr int BB = 2, CC = 3, HH = 96, WW = 96, DD = 64;
constexpr int NPX = HH * WW;            // 9216 pixels
constexpr int NT  = NPX / 16;           // 576 16-wide tiles per dim
constexpr int NPAIR = NT / 2;           // 288 balanced row-pair blocks
constexpr float THR = 0.7f;

// workspace layout (in floats)
constexpr size_t OFF_SN  = 0;                                    // B*N*D normalized semantics, pixel-major
constexpr size_t OFF_E   = OFF_SN  + (size_t)BB * NPX * DD;      // B*N*4 (e0,e1,e2,0)
constexpr size_t OFF_SSQ = OFF_E   + (size_t)BB * NPX * 4;       // B*N
constexpr size_t OFF_ACC = OFF_SSQ + (size_t)BB * NPX;           // B*4: cnt, SS, cross, pad
}

// ---------------------------------------------------------------------------
// Prep: normalize semantics into pixel-major layout, build e and ssq, zero acc
// ---------------------------------------------------------------------------
__global__ void __launch_bounds__(256)
cc_prep_kernel(const float* __restrict__ pred,
               const float* __restrict__ targ,
               const float* __restrict__ sem,
               float* __restrict__ ws) {
    const int idx = blockIdx.x * blockDim.x + threadIdx.x;
    if (blockIdx.x == 0 && threadIdx.x < BB * 4) ws[OFF_ACC + threadIdx.x] = 0.0f;
    if (idx >= BB * NPX) return;
    const int b = idx / NPX;
    const int n = idx - b * NPX;

    // load 64 channel-strided values (coalesced across lanes), normalize
    const float* s = sem + (size_t)b * DD * NPX + n;
    float sv[DD];
    float ss = 0.0f;
#pragma unroll
    for (int d = 0; d < DD; ++d) {
        float v = s[(size_t)d * NPX];
        sv[d] = v;
        ss += v * v;
    }
    const float inv = 1.0f / fmaxf(sqrtf(ss), 1e-12f);
    float* snp = ws + OFF_SN + (size_t)idx * DD;
#pragma unroll
    for (int d = 0; d < DD; d += 4) {
        float4 o = make_float4(sv[d] * inv, sv[d + 1] * inv, sv[d + 2] * inv, sv[d + 3] * inv);
        *(float4*)(snp + d) = o;
    }

    // e = pred - target (3 channels, padded to 4), ssq = |e|^2
    const float* p = pred + (size_t)b * CC * NPX + n;
    const float* t = targ + (size_t)b * CC * NPX + n;
    const float e0 = p[0]                 - t[0];
    const float e1 = p[(size_t)NPX]       - t[(size_t)NPX];
    const float e2 = p[(size_t)2 * NPX]   - t[(size_t)2 * NPX];
    *(float4*)(ws + OFF_E + (size_t)idx * 4) = make_float4(e0, e1, e2, 0.0f);
    ws[OFF_SSQ + idx] = e0 * e0 + e1 * e1 + e2 * e2;
}

// ---------------------------------------------------------------------------
// Main: tiled Gram matrix via V_WMMA_F32_16X16X4_F32 + fused threshold-reduce
// ---------------------------------------------------------------------------
__device__ __forceinline__ v8f wmma_f32x4(v2f a, v2f b, v8f c) {
    // D = A(16x4,f32) * B(4x16,f32) + C(16x16,f32)
    return __builtin_amdgcn_wmma_f32_16x16x4_f32(
        /*neg_a=*/false, a, /*neg_b=*/false, b,
        /*c_mod=*/(short)0, c, /*reuse_a=*/false, /*reuse_b=*/false);
}

__global__ void __launch_bounds__(256)
cc_sim_kernel(const float* __restrict__ ws_sn,
              const float* __restrict__ ws_e,
              const float* __restrict__ ws_q,
              float* __restrict__ acc) {
    const int b    = blockIdx.y;
    const int wave = threadIdx.x >> 5;
    const int lane = threadIdx.x & 31;
    const int l15  = lane & 15;
    const int hi   = lane >> 4;          // selects K pair {0,1} vs {2,3}

    const float* sn = ws_sn + (size_t)b * NPX * DD;
    const float* ef = ws_e  + (size_t)b * NPX * 4;
    const float* qq = ws_q  + (size_t)b * NPX;

    float c_cnt = 0.0f, c_ss = 0.0f, c_cr = 0.0f;

    // each block handles two row-strips (i and NT-1-i): constant 577 tiles/block
    for (int strip = 0; strip < 2; ++strip) {
        const int it = strip ? (NT - 1 - (int)blockIdx.x) : (int)blockIdx.x;
        const int i0 = it * 16;

        // A tile: rows i0..i0+15, full K=64, held in 32 VGPRs for the strip
        v2f a[16];
        const float* arow = sn + (size_t)(i0 + l15) * DD + 2 * hi;
#pragma unroll
        for (int kk = 0; kk < 16; ++kk)
            a[kk] = *(const v2f*)(arow + kk * 4);
        const v2f ae = *(const v2f*)(ef + (size_t)(i0 + l15) * 4 + 2 * hi);

        // per-row ssq for this strip (C/D layout: row m = v + 8*hi)
        float qr[8];
#pragma unroll
        for (int v = 0; v < 8; ++v) qr[v] = qq[i0 + v + 8 * hi];

        // 8 waves stride across the upper-triangular j tiles
        for (int jt = it + wave; jt < NT; jt += 8) {
            const int j0 = jt * 16;
            const float* brow = sn + (size_t)(j0 + l15) * DD + 2 * hi;

            v8f s = {};
#pragma unroll
            for (int kk = 0; kk < 16; ++kk) {
                v2f bb = *(const v2f*)(brow + kk * 4);
                s = wmma_f32x4(a[kk], bb, s);      // sim tile accumulate over K
            }
            const v2f be = *(const v2f*)(ef + (size_t)(j0 + l15) * 4 + 2 * hi);
            v8f zc = {};
            v8f eg = wmma_f32x4(ae, be, zc);       // <e_i, e_j> tile (K=4, padded)

            const float qc = qq[j0 + l15];         // ssq of column n = lane&15
            const float w  = (jt == it) ? 1.0f : 2.0f;  // symmetry weight
#pragma unroll
            for (int v = 0; v < 8; ++v) {
                const float m = (s[v] > THR) ? w : 0.0f;
                c_cnt += m;
                c_ss  += m * (qr[v] + qc);
                c_cr  += m * eg[v];
            }
        }
    }

    // wave32 butterfly reduction
#pragma unroll
    for (int off = 16; off > 0; off >>= 1) {
        c_cnt += __shfl_xor(c_cnt, off, 32);
        c_ss  += __shfl_xor(c_ss,  off, 32);
        c_cr  += __shfl_xor(c_cr,  off, 32);
    }
    if (lane == 0) {
        atomicAdd(&acc[b * 4 + 0], c_cnt);
        atomicAdd(&acc[b * 4 + 1], c_ss);
        atomicAdd(&acc[b * 4 + 2], c_cr);
    }
}

// ---------------------------------------------------------------------------
// Finalize: loss = sum_b (SS_b - 2*cross_b) / (cnt_b * C) / B
// ---------------------------------------------------------------------------
__global__ void cc_finalize_kernel(const float* __restrict__ acc, float* __restrict__ out) {
    if (blockIdx.x == 0 && threadIdx.x == 0) {
        float tot = 0.0f;
#pragma unroll
        for (int b = 0; b < BB; ++b) {
            const float cnt = acc[b * 4 + 0];
            const float ss  = acc[b * 4 + 1];
            const float cr  = acc[b * 4 + 2];
            const float pl  = (ss - 2.0f * cr) / (cnt * (float)CC);
            tot += (cnt > 0.0f) ? pl : 0.0f;
        }
        out[0] = tot / (float)BB;
    }
}

extern "C" void kernel_launch(void* const* d_in, const int* in_sizes, int n_in,
                              void* d_out, int out_size, void* d_ws, size_t ws_size,
                              hipStream_t stream) {
    (void)in_sizes; (void)n_in; (void)out_size; (void)ws_size;
    const float* pred = (const float*)d_in[0];
    const float* targ = (const float*)d_in[1];
    const float* sem  = (const float*)d_in[2];
    float* ws  = (float*)d_ws;
    float* out = (float*)d_out;

    const int prep_threads = BB * NPX;
    cc_prep_kernel<<<(prep_threads + 255) / 256, 256, 0, stream>>>(pred, targ, sem, ws);

    dim3 grid(NPAIR, BB);
    cc_sim_kernel<<<grid, 256, 0, stream>>>(ws + OFF_SN, ws + OFF_E, ws + OFF_SSQ, ws + OFF_ACC);

    cc_finalize_kernel<<<1, 32, 0, stream>>>(ws + OFF_ACC, out);
}